// FlowNet3D_953482739749
// MI455X (gfx1250) — compile-verified
//
#include <hip/hip_runtime.h>

typedef __attribute__((ext_vector_type(2))) float v2f;
typedef __attribute__((ext_vector_type(8))) float v8f;

#define GRID_STRIDE(t, total)                                                  \
  for (size_t t = blockIdx.x * (size_t)blockDim.x + threadIdx.x; t < (total);  \
       t += (size_t)gridDim.x * blockDim.x)

// ---------------------------------------------------------------------------
// transpose (B,3,N) -> (B,N,3)
// ---------------------------------------------------------------------------
__global__ void k_transpose(const float* __restrict__ in, float* __restrict__ out,
                            int B, int N) {
  size_t total = (size_t)B * N * 3;
  GRID_STRIDE(t, total) {
    int c = (int)(t % 3);
    size_t r = t / 3;
    int n = (int)(r % N);
    int b = (int)(r / N);
    out[t] = in[((size_t)b * 3 + c) * N + n];
  }
}

// ---------------------------------------------------------------------------
// Farthest point sampling: one 1024-thread block per batch.
// dist state in registers (<=8 points/thread), wave32 shuffle reduction,
// first-index tie-break to match jnp.argmax.
// ---------------------------------------------------------------------------
__global__ void k_fps(const float* __restrict__ xyz, int N, int npoint,
                      int* __restrict__ out_idx, float* __restrict__ new_xyz) {
  const int b = blockIdx.x;
  const int t = threadIdx.x;
  const float* P = xyz + (size_t)b * N * 3;
  const int np = (N + 1023) >> 10;
  float dist[8];
#pragma unroll
  for (int i = 0; i < 8; ++i) dist[i] = 1e10f;

  __shared__ float swd[32];
  __shared__ int swi[32];
  __shared__ float sf[3];
  __shared__ int sfar;

  int far = 0;
  for (int s = 0; s < npoint; ++s) {
    if (t == 0) {
      out_idx[(size_t)b * npoint + s] = far;
      float fx = P[(size_t)far * 3 + 0];
      float fy = P[(size_t)far * 3 + 1];
      float fz = P[(size_t)far * 3 + 2];
      sf[0] = fx; sf[1] = fy; sf[2] = fz;
      size_t o = ((size_t)b * npoint + s) * 3;
      new_xyz[o + 0] = fx; new_xyz[o + 1] = fy; new_xyz[o + 2] = fz;
    }
    __syncthreads();
    float fx = sf[0], fy = sf[1], fz = sf[2];
    float bd = -1.0f; int bi = 0;
    for (int i = 0; i < np; ++i) {
      int p = t + (i << 10);
      if (p < N) {
        float dx = P[(size_t)p * 3 + 0] - fx;
        float dy = P[(size_t)p * 3 + 1] - fy;
        float dz = P[(size_t)p * 3 + 2] - fz;
        float d = dx * dx + dy * dy + dz * dz;
        float cur = fminf(dist[i], d);
        dist[i] = cur;
        if (cur > bd || (cur == bd && p < bi)) { bd = cur; bi = p; }
      }
    }
#pragma unroll
    for (int off = 16; off > 0; off >>= 1) {
      float od = __shfl_xor(bd, off, 32);
      int oi = __shfl_xor(bi, off, 32);
      if (od > bd || (od == bd && oi < bi)) { bd = od; bi = oi; }
    }
    if ((t & 31) == 0) { swd[t >> 5] = bd; swi[t >> 5] = bi; }
    __syncthreads();
    if (t < 32) {
      int nw = blockDim.x >> 5;
      float rd = (t < nw) ? swd[t] : -1.0f;
      int ri = (t < nw) ? swi[t] : 0;
#pragma unroll
      for (int off = 16; off > 0; off >>= 1) {
        float od = __shfl_xor(rd, off, 32);
        int oi = __shfl_xor(ri, off, 32);
        if (od > rd || (od == rd && oi < ri)) { rd = od; ri = oi; }
      }
      if (t == 0) sfar = ri;
    }
    __syncthreads();
    far = sfar;
  }
}

// ---------------------------------------------------------------------------
// Ball query: first nsample indices (ascending) with d<=r^2, pad with first.
// ---------------------------------------------------------------------------
__global__ void k_ball(const float* __restrict__ xyz, const float* __restrict__ ctr,
                       int* __restrict__ idx, int B, int N, int npoint, int ns,
                       float r2) {
  int g = blockIdx.x * blockDim.x + threadIdx.x;
  if (g >= B * npoint) return;
  int b = g / npoint;
  const float* P = xyz + (size_t)b * N * 3;
  float cx = ctr[(size_t)g * 3 + 0];
  float cy = ctr[(size_t)g * 3 + 1];
  float cz = ctr[(size_t)g * 3 + 2];
  int* out = idx + (size_t)g * ns;
  int cnt = 0, first = N - 1;
  for (int p = 0; p < N && cnt < ns; ++p) {
    float dx = P[(size_t)p * 3 + 0] - cx;
    float dy = P[(size_t)p * 3 + 1] - cy;
    float dz = P[(size_t)p * 3 + 2] - cz;
    if (dx * dx + dy * dy + dz * dz <= r2) {
      if (cnt == 0) first = p;
      out[cnt++] = p;
    }
  }
  for (int k = cnt; k < ns; ++k) out[k] = first;
}

// ---------------------------------------------------------------------------
// kNN via iterative (d, idx)-lexicographic min extraction (matches top_k order)
// ---------------------------------------------------------------------------
__global__ void k_knn(const float* __restrict__ q, const float* __restrict__ ref,
                      int B, int Nq, int Nr, int K, int* __restrict__ idx,
                      float* __restrict__ dst) {
  int g = blockIdx.x * blockDim.x + threadIdx.x;
  if (g >= B * Nq) return;
  int b = g / Nq;
  const float* R = ref + (size_t)b * Nr * 3;
  float qx = q[(size_t)g * 3 + 0];
  float qy = q[(size_t)g * 3 + 1];
  float qz = q[(size_t)g * 3 + 2];
  float lastd = -1e30f; int lasti = -1;
  for (int k = 0; k < K; ++k) {
    float bd = 1e30f; int bi = Nr;
    for (int p = 0; p < Nr; ++p) {
      float dx = R[(size_t)p * 3 + 0] - qx;
      float dy = R[(size_t)p * 3 + 1] - qy;
      float dz = R[(size_t)p * 3 + 2] - qz;
      float d = dx * dx + dy * dy + dz * dz;
      bool gtLast = (d > lastd) || (d == lastd && p > lasti);
      bool ltBest = (d < bd) || (d == bd && p < bi);
      if (gtLast && ltBest) { bd = d; bi = p; }
    }
    idx[(size_t)g * K + k] = bi;
    if (dst) dst[(size_t)g * K + k] = bd;
    lastd = bd; lasti = bi;
  }
}

// ---------------------------------------------------------------------------
// Gathers / concat / rowmax / interpolation
// ---------------------------------------------------------------------------
__global__ void k_sa_gather(const float* __restrict__ xyz, const float* __restrict__ feat,
                            const float* __restrict__ ctr, const int* __restrict__ idx,
                            float* __restrict__ X, int B, int N, int npoint, int ns, int C) {
  int Cin = 3 + C;
  size_t total = (size_t)B * npoint * ns * Cin;
  GRID_STRIDE(t, total) {
    int c = (int)(t % Cin);
    size_t r = t / Cin;
    int s = (int)(r % ns);
    size_t g = r / ns;               // b*npoint + j
    int b = (int)(g / npoint);
    int p = idx[g * ns + s];
    float v;
    if (c < 3) v = xyz[((size_t)b * N + p) * 3 + c] - ctr[g * 3 + c];
    else       v = feat[((size_t)b * N + p) * C + (c - 3)];
    X[t] = v;
  }
}

// flow-embed gather, single batch: [pos_diff(3), feat2[idx](C), feat1(C)]
__global__ void k_fe_gather(const float* __restrict__ pos1, const float* __restrict__ pos2,
                            const float* __restrict__ feat1, const float* __restrict__ feat2,
                            const int* __restrict__ idx, float* __restrict__ X,
                            int N1, int N2, int C, int K) {
  int Cin = 3 + 2 * C;
  size_t total = (size_t)N1 * K * Cin;
  GRID_STRIDE(t, total) {
    int c = (int)(t % Cin);
    size_t r = t / Cin;
    int s = (int)(r % K);
    int n = (int)(r / K);
    int p = idx[(size_t)n * K + s];
    float v;
    if (c < 3)          v = pos2[(size_t)p * 3 + c] - pos1[(size_t)n * 3 + c];
    else if (c < 3 + C) v = feat2[(size_t)p * C + (c - 3)];
    else                v = feat1[(size_t)n * C + (c - 3 - C)];
    X[t] = v;
  }
}

// upconv gather: [feat2[idx](C2), pos_diff(3)]
__global__ void k_su_gather(const float* __restrict__ pos1, const float* __restrict__ pos2,
                            const float* __restrict__ feat2, const int* __restrict__ idx,
                            float* __restrict__ X, int B, int N1, int N2, int C2, int K) {
  int Cin = C2 + 3;
  size_t total = (size_t)B * N1 * K * Cin;
  GRID_STRIDE(t, total) {
    int c = (int)(t % Cin);
    size_t r = t / Cin;
    int s = (int)(r % K);
    size_t qi = r / K;               // b*N1 + n
    int b = (int)(qi / N1);
    int p = idx[qi * K + s];
    float v;
    if (c < C2) v = feat2[((size_t)b * N2 + p) * C2 + c];
    else {
      int cc = c - C2;
      v = pos2[((size_t)b * N2 + p) * 3 + cc] - pos1[qi * 3 + cc];
    }
    X[t] = v;
  }
}

__global__ void k_concat(const float* __restrict__ A, int Ca,
                         const float* __restrict__ Bm, int Cb,
                         float* __restrict__ Y, size_t rows) {
  int Cin = Ca + Cb;
  size_t total = rows * Cin;
  GRID_STRIDE(t, total) {
    int c = (int)(t % Cin);
    size_t r = t / Cin;
    Y[t] = (c < Ca) ? A[r * Ca + c] : Bm[r * Cb + (c - Ca)];
  }
}

__global__ void k_rowmax(const float* __restrict__ X, float* __restrict__ Y,
                         size_t G, int S, int C) {
  size_t total = G * C;
  GRID_STRIDE(t, total) {
    int c = (int)(t % C);
    size_t g = t / C;
    float m = -3.4e38f;
    for (int s = 0; s < S; ++s) m = fmaxf(m, X[(g * S + s) * C + c]);
    Y[t] = m;
  }
}

// feature propagation: inverse-distance interp of feat2 (C2) ++ feat1 (C1)
__global__ void k_fp_interp(const float* __restrict__ feat2, const float* __restrict__ feat1,
                            const int* __restrict__ idx, const float* __restrict__ dst,
                            float* __restrict__ X, int B, int N1, int N2, int C2, int C1) {
  int Cin = C2 + C1;
  size_t total = (size_t)B * N1 * Cin;
  GRID_STRIDE(t, total) {
    int c = (int)(t % Cin);
    size_t r = t / Cin;              // b*N1 + n
    int b = (int)(r / N1);
    if (c < C2) {
      const int* id = idx + r * 3;
      const float* dd = dst + r * 3;
      float w0 = 1.0f / fmaxf(dd[0], 1e-10f);
      float w1 = 1.0f / fmaxf(dd[1], 1e-10f);
      float w2 = 1.0f / fmaxf(dd[2], 1e-10f);
      float inv = 1.0f / (w0 + w1 + w2);
      float v = w0 * feat2[((size_t)b * N2 + id[0]) * C2 + c] +
                w1 * feat2[((size_t)b * N2 + id[1]) * C2 + c] +
                w2 * feat2[((size_t)b * N2 + id[2]) * C2 + c];
      X[t] = v * inv;
    } else {
      X[t] = feat1[r * C1 + (c - C2)];
    }
  }
}

// ---------------------------------------------------------------------------
// fp32 WMMA GEMM: Y = act(X @ W^T + bias)
//   X: (M,K) row-major (M % 16 == 0)   W: (N,K) row-major   Y: (M,N)
// One wave computes a 16(M) x 32(N) tile: A fragment reused by two B tiles
// (2 v_wmma_f32_16x16x4_f32 per 4-wide K step). Main K loop is branchless:
// rows are always valid; out-of-range columns clamp their W pointer to row 0
// (garbage only feeds never-written output columns). Single guarded tail
// handles K % 4.
// ---------------------------------------------------------------------------
__global__ void __launch_bounds__(32)
k_gemm_wmma(const float* __restrict__ X, const float* __restrict__ W,
            const float* __restrict__ bias, float* __restrict__ Y,
            int M, int N, int K, int relu) {
  const int lane = threadIdx.x;
  const int half = lane >> 4;
  const int l15 = lane & 15;
  const int m0 = blockIdx.y * 16;
  const int n0 = blockIdx.x * 32;
  const int row = m0 + l15;
  const int col0 = n0 + l15;
  const int col1 = n0 + 16 + l15;
  const bool c0ok = (col0 < N);
  const bool c1ok = (col1 < N);
  // element (k0 + 2*half) of each stream == xr[k0]
  const float* __restrict__ xr = X + (size_t)row * K + 2 * half;
  const float* __restrict__ w0r = W + (size_t)(c0ok ? col0 : 0) * K + 2 * half;
  const float* __restrict__ w1r = W + (size_t)(c1ok ? col1 : 0) * K + 2 * half;

  __builtin_prefetch(xr, 0, 0);
  __builtin_prefetch(w0r, 0, 0);
  __builtin_prefetch(w1r, 0, 0);

  v8f acc0 = {};
  v8f acc1 = {};
  const int Km = K & ~3;
#pragma unroll 2
  for (int k0 = 0; k0 < Km; k0 += 4) {
    v2f a, b0, b1;
    a.x  = xr[k0];
    a.y  = xr[k0 + 1];
    b0.x = w0r[k0];
    b0.y = w0r[k0 + 1];
    b1.x = w1r[k0];
    b1.y = w1r[k0 + 1];
    acc0 = __builtin_amdgcn_wmma_f32_16x16x4_f32(false, a, false, b0, (short)0,
                                                 acc0, false, false);
    acc1 = __builtin_amdgcn_wmma_f32_16x16x4_f32(false, a, false, b1, (short)0,
                                                 acc1, false, false);
  }
  if (Km < K) {  // guarded tail for K % 4 (runs once)
    const int ka = Km + 2 * half;
    const bool e0 = (ka < K);
    const bool e1 = (ka + 1 < K);
    v2f a, b0, b1;
    a.x  = e0 ? xr[Km] : 0.0f;
    a.y  = e1 ? xr[Km + 1] : 0.0f;
    b0.x = e0 ? w0r[Km] : 0.0f;
    b0.y = e1 ? w0r[Km + 1] : 0.0f;
    b1.x = e0 ? w1r[Km] : 0.0f;
    b1.y = e1 ? w1r[Km + 1] : 0.0f;
    acc0 = __builtin_amdgcn_wmma_f32_16x16x4_f32(false, a, false, b0, (short)0,
                                                 acc0, false, false);
    acc1 = __builtin_amdgcn_wmma_f32_16x16x4_f32(false, a, false, b1, (short)0,
                                                 acc1, false, false);
  }

  const int mb = m0 + half * 8;
  if (c0ok) {
    float bv = bias ? bias[col0] : 0.0f;
#pragma unroll
    for (int v = 0; v < 8; ++v) {
      float val = acc0[v] + bv;
      if (relu) val = fmaxf(val, 0.0f);
      Y[(size_t)(mb + v) * N + col0] = val;
    }
  }
  if (c1ok) {
    float bv = bias ? bias[col1] : 0.0f;
#pragma unroll
    for (int v = 0; v < 8; ++v) {
      float val = acc1[v] + bv;
      if (relu) val = fmaxf(val, 0.0f);
      Y[(size_t)(mb + v) * N + col1] = val;
    }
  }
}

// ---------------------------------------------------------------------------
// Host-side helpers
// ---------------------------------------------------------------------------
static inline int gsBlocks(size_t total) {
  size_t b = (total + 255) / 256;
  if (b > (size_t)1048576) b = 1048576;
  if (b < 1) b = 1;
  return (int)b;
}

static void gemm(hipStream_t s, const float* X, const float* W, const float* bias,
                 float* Y, int M, int N, int K, bool relu) {
  dim3 grid((unsigned)((N + 31) / 32), (unsigned)(M / 16));
  k_gemm_wmma<<<grid, dim3(32), 0, s>>>(X, W, bias, Y, M, N, K, relu ? 1 : 0);
}

static void run_sa(hipStream_t s, const float* xyz, const float* feat, int B, int N,
                   int C, int npoint, int ns, float radius, const float* W0, int o0,
                   const float* W1, int o1, const float* W2, int o2, float* new_xyz,
                   float* new_feat, int* fpsIdx, int* ballIdx, float* bufX, float* H1,
                   float* H2) {
  k_fps<<<B, 1024, 0, s>>>(xyz, N, npoint, fpsIdx, new_xyz);
  int nc = B * npoint;
  k_ball<<<(nc + 255) / 256, 256, 0, s>>>(xyz, new_xyz, ballIdx, B, N, npoint, ns,
                                          radius * radius);
  int Cin = 3 + C;
  size_t rows = (size_t)nc * ns;
  k_sa_gather<<<gsBlocks(rows * Cin), 256, 0, s>>>(xyz, feat, new_xyz, ballIdx, bufX,
                                                   B, N, npoint, ns, C);
  gemm(s, bufX, W0, nullptr, H1, (int)rows, o0, Cin, true);
  gemm(s, H1, W1, nullptr, H2, (int)rows, o1, o0, true);
  gemm(s, H2, W2, nullptr, H1, (int)rows, o2, o1, true);
  k_rowmax<<<gsBlocks((size_t)nc * o2), 256, 0, s>>>(H1, new_feat, (size_t)nc, ns, o2);
}

extern "C" void kernel_launch(void* const* d_in, const int* in_sizes, int n_in,
                              void* d_out, int out_size, void* d_ws, size_t ws_size,
                              hipStream_t stream) {
  const int B = 2, N0 = 8192;
  const float* pc1 = (const float*)d_in[0];
  const float* pc2 = (const float*)d_in[1];
  const float* fe1 = (const float*)d_in[2];
  const float* fe2 = (const float*)d_in[3];
  const float* sa1w0 = (const float*)d_in[4];
  const float* sa1w1 = (const float*)d_in[5];
  const float* sa1w2 = (const float*)d_in[6];
  const float* sa2w0 = (const float*)d_in[7];
  const float* sa2w1 = (const float*)d_in[8];
  const float* sa2w2 = (const float*)d_in[9];
  const float* sa3w0 = (const float*)d_in[10];
  const float* sa3w1 = (const float*)d_in[11];
  const float* sa3w2 = (const float*)d_in[12];
  const float* sa4w0 = (const float*)d_in[13];
  const float* sa4w1 = (const float*)d_in[14];
  const float* sa4w2 = (const float*)d_in[15];
  const float* few0 = (const float*)d_in[16];
  const float* few1 = (const float*)d_in[17];
  const float* few2 = (const float*)d_in[18];
  const float* su1b0 = (const float*)d_in[19];   // (256,771)
  const float* su1b1 = (const float*)d_in[20];   // (256,256)
  const float* su2m0 = (const float*)d_in[21];   // (128,259)
  const float* su2m1 = (const float*)d_in[22];   // (128,128)
  const float* su2m2 = (const float*)d_in[23];   // (256,128)
  const float* su2b0 = (const float*)d_in[24];   // (256,512)
  const float* su3m0 = (const float*)d_in[25];
  const float* su3m1 = (const float*)d_in[26];
  const float* su3m2 = (const float*)d_in[27];
  const float* su3b0 = (const float*)d_in[28];   // (256,320)
  const float* fpw0 = (const float*)d_in[29];    // (256,259)
  const float* fpw1 = (const float*)d_in[30];    // (256,256)
  const float* conv1 = (const float*)d_in[31];   // (128,256)
  const float* conv2w = (const float*)d_in[32];  // (3,128)
  const float* conv2b = (const float*)d_in[33];  // (3,)

  // bump allocator over d_ws
  char* base = (char*)d_ws;
  size_t off = 0;
  auto allocF = [&](size_t n) -> float* {
    float* p = (float*)(base + off);
    off = (off + n * sizeof(float) + 255) & ~(size_t)255;
    return p;
  };
  auto allocI = [&](size_t n) -> int* {
    int* p = (int*)(base + off);
    off = (off + n * sizeof(int) + 255) & ~(size_t)255;
    return p;
  };

  float* p1 = allocF(49152);   float* p2 = allocF(49152);
  float* f1 = allocF(49152);   float* f2 = allocF(49152);
  float* l1p_a = allocF(24576);  float* l1f_a = allocF(524288);
  float* l1p_b = allocF(24576);  float* l1f_b = allocF(524288);
  float* l2p_a = allocF(6144);   float* l2f_a = allocF(262144);
  float* l2p_b = allocF(6144);   float* l2f_b = allocF(262144);
  float* l2fnew = allocF(262144);
  float* l3p = allocF(1536);     float* l3f = allocF(131072);
  float* l4p = allocF(384);      float* l4f = allocF(65536);
  float* l3fn = allocF(131072);
  float* l2fn = allocF(524288);
  float* l1fn = allocF(2097152);
  float* l0fn = allocF(4194304);
  float* featcat = allocF(524288);
  float* premax = allocF(2097152);    // also reused as conv hidden (16384x128)
  float* catbuf = allocF(4243456);
  float* bufX = allocF((size_t)65536 * 260);
  float* H1 = allocF((size_t)65536 * 256);
  float* H2 = allocF((size_t)65536 * 256);
  int* fpsIdx = allocI(8192);
  int* ballIdx = allocI(131072);
  int* knnIdx = allocI(131072);
  float* knnDst = allocF(49152);
  (void)ws_size; (void)n_in; (void)in_sizes; (void)out_size;

  // ---- transpose inputs (B,3,N) -> (B,N,3)
  k_transpose<<<gsBlocks(49152), 256, 0, stream>>>(pc1, p1, B, N0);
  k_transpose<<<gsBlocks(49152), 256, 0, stream>>>(pc2, p2, B, N0);
  k_transpose<<<gsBlocks(49152), 256, 0, stream>>>(fe1, f1, B, N0);
  k_transpose<<<gsBlocks(49152), 256, 0, stream>>>(fe2, f2, B, N0);

  // ---- SA1 / SA2 for both clouds
  run_sa(stream, p1, f1, B, 8192, 3, 4096, 16, 0.02f, sa1w0, 32, sa1w1, 32, sa1w2, 64,
         l1p_a, l1f_a, fpsIdx, ballIdx, bufX, H1, H2);
  run_sa(stream, l1p_a, l1f_a, B, 4096, 64, 1024, 16, 0.04f, sa2w0, 64, sa2w1, 64,
         sa2w2, 128, l2p_a, l2f_a, fpsIdx, ballIdx, bufX, H1, H2);
  run_sa(stream, p2, f2, B, 8192, 3, 4096, 16, 0.02f, sa1w0, 32, sa1w1, 32, sa1w2, 64,
         l1p_b, l1f_b, fpsIdx, ballIdx, bufX, H1, H2);
  run_sa(stream, l1p_b, l1f_b, B, 4096, 64, 1024, 16, 0.04f, sa2w0, 64, sa2w1, 64,
         sa2w2, 128, l2p_b, l2f_b, fpsIdx, ballIdx, bufX, H1, H2);

  // ---- Flow embedding (K=64), per batch to bound scratch
  for (int b = 0; b < B; ++b) {
    const float* q = l2p_a + (size_t)b * 1024 * 3;
    const float* r = l2p_b + (size_t)b * 1024 * 3;
    const float* fq = l2f_a + (size_t)b * 1024 * 128;
    const float* fr = l2f_b + (size_t)b * 1024 * 128;
    k_knn<<<(1024 + 255) / 256, 256, 0, stream>>>(q, r, 1, 1024, 1024, 64, knnIdx,
                                                  nullptr);
    size_t rows = (size_t)1024 * 64;
    k_fe_gather<<<gsBlocks(rows * 259), 256, 0, stream>>>(q, r, fq, fr, knnIdx, bufX,
                                                          1024, 1024, 128, 64);
    gemm(stream, bufX, few0, nullptr, H1, (int)rows, 128, 259, true);
    gemm(stream, H1, few1, nullptr, H2, (int)rows, 128, 128, true);
    gemm(stream, H2, few2, nullptr, H1, (int)rows, 128, 128, true);
    k_rowmax<<<gsBlocks((size_t)1024 * 128), 256, 0, stream>>>(
        H1, l2fnew + (size_t)b * 1024 * 128, (size_t)1024, 64, 128);
  }

  // ---- SA3 / SA4 on cloud-1 path
  run_sa(stream, l2p_a, l2fnew, B, 1024, 128, 256, 8, 0.08f, sa3w0, 128, sa3w1, 128,
         sa3w2, 256, l3p, l3f, fpsIdx, ballIdx, bufX, H1, H2);
  run_sa(stream, l3p, l3f, B, 256, 256, 64, 8, 0.16f, sa4w0, 256, sa4w1, 256, sa4w2,
         512, l4p, l4f, fpsIdx, ballIdx, bufX, H1, H2);

  // ---- su1: l3 <- l4 (no mlp1; raw max of [feat2(512), posdiff(3)])
  k_knn<<<(512 + 255) / 256, 256, 0, stream>>>(l3p, l4p, B, 256, 64, 8, knnIdx,
                                               nullptr);
  {
    size_t rows = (size_t)B * 256 * 8;  // 4096
    k_su_gather<<<gsBlocks(rows * 515), 256, 0, stream>>>(l3p, l4p, l4f, knnIdx, bufX,
                                                          B, 256, 64, 512, 8);
    k_rowmax<<<gsBlocks((size_t)512 * 515), 256, 0, stream>>>(bufX, premax,
                                                              (size_t)512, 8, 515);
    k_concat<<<gsBlocks((size_t)512 * 771), 256, 0, stream>>>(premax, 515, l3f, 256,
                                                              catbuf, (size_t)512);
    gemm(stream, catbuf, su1b0, nullptr, H1, 512, 256, 771, true);
    gemm(stream, H1, su1b1, nullptr, l3fn, 512, 256, 256, true);
  }

  // ---- featcat = [l2f_a(128), l2fnew(128)]
  k_concat<<<gsBlocks((size_t)2048 * 256), 256, 0, stream>>>(l2f_a, 128, l2fnew, 128,
                                                             featcat, (size_t)2048);

  // ---- su2: l2 <- l3
  k_knn<<<(2048 + 255) / 256, 256, 0, stream>>>(l2p_a, l3p, B, 1024, 256, 8, knnIdx,
                                                nullptr);
  {
    size_t rows = (size_t)B * 1024 * 8;  // 16384
    k_su_gather<<<gsBlocks(rows * 259), 256, 0, stream>>>(l2p_a, l3p, l3fn, knnIdx,
                                                          bufX, B, 1024, 256, 256, 8);
    gemm(stream, bufX, su2m0, nullptr, H1, (int)rows, 128, 259, true);
    gemm(stream, H1, su2m1, nullptr, H2, (int)rows, 128, 128, true);
    gemm(stream, H2, su2m2, nullptr, H1, (int)rows, 256, 128, true);
    k_rowmax<<<gsBlocks((size_t)2048 * 256), 256, 0, stream>>>(H1, premax,
                                                               (size_t)2048, 8, 256);
    k_concat<<<gsBlocks((size_t)2048 * 512), 256, 0, stream>>>(premax, 256, featcat,
                                                               256, catbuf,
                                                               (size_t)2048);
    gemm(stream, catbuf, su2b0, nullptr, l2fn, 2048, 256, 512, true);
  }

  // ---- su3: l1 <- l2
  k_knn<<<(8192 + 255) / 256, 256, 0, stream>>>(l1p_a, l2p_a, B, 4096, 1024, 8,
                                                knnIdx, nullptr);
  {
    size_t rows = (size_t)B * 4096 * 8;  // 65536
    k_su_gather<<<gsBlocks(rows * 259), 256, 0, stream>>>(l1p_a, l2p_a, l2fn, knnIdx,
                                                          bufX, B, 4096, 1024, 256, 8);
    gemm(stream, bufX, su3m0, nullptr, H1, (int)rows, 128, 259, true);
    gemm(stream, H1, su3m1, nullptr, H2, (int)rows, 128, 128, true);
    gemm(stream, H2, su3m2, nullptr, H1, (int)rows, 256, 128, true);
    k_rowmax<<<gsBlocks((size_t)8192 * 256), 256, 0, stream>>>(H1, premax,
                                                               (size_t)8192, 8, 256);
    k_concat<<<gsBlocks((size_t)8192 * 320), 256, 0, stream>>>(premax, 256, l1f_a, 64,
                                                               catbuf, (size_t)8192);
    gemm(stream, catbuf, su3b0, nullptr, l1fn, 8192, 256, 320, true);
  }

  // ---- feature propagation: l0 <- l1 (K=3 inverse-distance)
  k_knn<<<(16384 + 255) / 256, 256, 0, stream>>>(p1, l1p_a, B, 8192, 4096, 3, knnIdx,
                                                 knnDst);
  k_fp_interp<<<gsBlocks((size_t)16384 * 259), 256, 0, stream>>>(
      l1fn, f1, knnIdx, knnDst, catbuf, B, 8192, 4096, 256, 3);
  gemm(stream, catbuf, fpw0, nullptr, H1, 16384, 256, 259, true);
  gemm(stream, H1, fpw1, nullptr, l0fn, 16384, 256, 256, true);

  // ---- final convs
  gemm(stream, l0fn, conv1, nullptr, premax, 16384, 128, 256, true);
  gemm(stream, premax, conv2w, conv2b, (float*)d_out, 16384, 3, 128, false);
}